// ZINBGAE_27118423507715
// MI455X (gfx1250) — compile-verified
//
#include <hip/hip_runtime.h>
#include <hip/hip_bf16.h>
#include <math.h>

#define NN    10000
#define FIN   512
#define HD    40
#define HP    48          // padded hidden for layer-1 N tiles
#define MT    625         // NN/16 exactly

typedef __attribute__((ext_vector_type(2))) float v2f;
typedef __attribute__((ext_vector_type(8))) float v8f;

__device__ __forceinline__ v8f wmma_f32x4(v2f a, v2f b, v8f c) {
#if defined(__gfx1250__) && __has_builtin(__builtin_amdgcn_wmma_f32_16x16x4_f32)
  // 8 args: (neg_a, A, neg_b, B, c_mod, C, reuse_a, reuse_b)
  return __builtin_amdgcn_wmma_f32_16x16x4_f32(false, a, false, b, (short)0, c,
                                               false, false);
#else
  return c; // never taken on gfx1250
#endif
}

// ---- init: deg=0, agg1[n][c]=b1[c], aggH=0 --------------------------------
__global__ void init_kernel(float* __restrict__ deg, float* __restrict__ agg1,
                            float* __restrict__ aggH, const float* __restrict__ b1) {
  int i = blockIdx.x * blockDim.x + threadIdx.x;
  if (i < NN * HD) {
    agg1[i] = b1[i % HD];
    aggH[i] = 0.0f;
  }
  if (i < NN) deg[i] = 0.0f;
}

// ---- weight prep: W1t[48][512] (transposed, zero-padded), Wht[3][512][40] -
__global__ void prep_weights(const float* __restrict__ W1,
                             const float* __restrict__ Wm, const float* __restrict__ Wd,
                             const float* __restrict__ Wp,
                             float* __restrict__ W1t, float* __restrict__ Wht) {
  int i = blockIdx.x * blockDim.x + threadIdx.x;
  if (i < HP * FIN) {                       // W1t[c][k] = (c<40) ? W1[k][c] : 0
    int c = i >> 9;                         // /512
    int k = i & (FIN - 1);
    W1t[i] = (c < HD) ? W1[k * HD + c] : 0.0f;
  }
  if (i < 3 * FIN * HD) {                   // Wht[h][n][k] = Wh[k][n]
    int head = i / (FIN * HD);
    int r    = i % (FIN * HD);
    int n = r / HD, k = r % HD;
    const float* W = (head == 0) ? Wm : (head == 1) ? Wd : Wp;
    Wht[i] = W[k * FIN + n];
  }
}

// ---- degree: deg[dst] += 1 ------------------------------------------------
__global__ void degree_kernel(const int* __restrict__ dst, float* __restrict__ deg, int E) {
  int e = blockIdx.x * blockDim.x + threadIdx.x;
  if (e < E) atomicAdd(&deg[dst[e]], 1.0f);
}

// ---- layer-1 GEMM: H1[10000,48] = X[10000,512] @ W1t^T --------------------
// blockDim = 96 (3 waves); wave w owns N-tile w (cols 16w..16w+15 of 48).
// All loads unconditional b64; padded cols 40..47 compute exact zeros.
__global__ void gemm1_wmma(const float* __restrict__ X, const float* __restrict__ W1t,
                           float* __restrict__ H1) {
  const int lane  = threadIdx.x & 31;
  const int warp  = threadIdx.x >> 5;   // 0..2 = nTile
  const int mTile = blockIdx.x;         // 0..624
  const int half  = lane >> 4;
  const int l16   = lane & 15;
  const long rowA = (long)(mTile * 16 + l16) * FIN;   // A row base
  const long colB = (long)(warp * 16 + l16) * FIN;    // B (transposed) row base

  v8f acc = {};
#pragma unroll 8
  for (int k0 = 0; k0 < FIN; k0 += 4) {
    const int ka = k0 + 2 * half;
    v2f a = *(const v2f*)(X   + rowA + ka);   // X[row][ka], X[row][ka+1]
    v2f b = *(const v2f*)(W1t + colB + ka);   // W1[ka][col], W1[ka+1][col]
    acc = wmma_f32x4(a, b, acc);
  }
  const int col = warp * 16 + l16;
#pragma unroll
  for (int v = 0; v < 8; ++v) {
    int m = mTile * 16 + v + 8 * half;
    H1[m * HP + col] = acc[v];
  }
}

// ---- edge scatter: out[dst][0:40] += h[src][0:40] * coef ------------------
__global__ void scatter_kernel(const float* __restrict__ h, const float* __restrict__ deg,
                               const int* __restrict__ src, const int* __restrict__ dst,
                               float* __restrict__ out, int E, int hstride) {
  int e = blockIdx.x * blockDim.x + threadIdx.x;
  if (e >= E) return;
  int s = src[e], d = dst[e];
  float coef = rsqrtf(deg[s]) * rsqrtf(deg[d]);
  const float4* hr4 = (const float4*)(h + (long)s * hstride);
  float* orow = out + (long)d * HD;
#pragma unroll
  for (int c4 = 0; c4 < HD / 4; ++c4) {
    float4 v = hr4[c4];
    atomicAdd(&orow[c4 * 4 + 0], v.x * coef);
    atomicAdd(&orow[c4 * 4 + 1], v.y * coef);
    atomicAdd(&orow[c4 * 4 + 2], v.z * coef);
    atomicAdd(&orow[c4 * 4 + 3], v.w * coef);
  }
}

// ---- GraphNorm stats: one block per channel -------------------------------
__global__ void gn_stats_kernel(const float* __restrict__ agg,
                                const float* __restrict__ gn_w, const float* __restrict__ gn_b,
                                const float* __restrict__ gn_a, float* __restrict__ stats) {
  const int c = blockIdx.x;           // 0..39
  __shared__ float s1[256], s2[256];
  float sum = 0.0f, sq = 0.0f;
  for (int n = threadIdx.x; n < NN; n += 256) {
    float v = agg[n * HD + c];
    sum += v; sq += v * v;
  }
  s1[threadIdx.x] = sum; s2[threadIdx.x] = sq;
  __syncthreads();
  for (int off = 128; off > 0; off >>= 1) {
    if (threadIdx.x < off) { s1[threadIdx.x] += s1[threadIdx.x + off];
                             s2[threadIdx.x] += s2[threadIdx.x + off]; }
    __syncthreads();
  }
  if (threadIdx.x == 0) {
    float mean = s1[0] * (1.0f / NN);
    float msq  = s2[0] * (1.0f / NN);
    float a = gn_a[c];
    // var = E[(x - a*mean)^2] = E[x^2] - 2a*mean^2 + a^2*mean^2
    float var = msq - 2.0f * a * mean * mean + a * a * mean * mean;
    float rstd = rsqrtf(var + 1e-5f);
    stats[c]          = gn_w[c] * rstd;  // scale
    stats[HD + c]     = a * mean;        // shift
    stats[2 * HD + c] = gn_b[c];         // bias
  }
}

// ---- normalize + ReLU (in place) ------------------------------------------
__global__ void norm_relu_kernel(float* __restrict__ agg, const float* __restrict__ stats) {
  int i = blockIdx.x * blockDim.x + threadIdx.x;
  if (i >= NN * HD) return;
  int c = i % HD;
  float v = (agg[i] - stats[HD + c]) * stats[c] + stats[2 * HD + c];
  agg[i] = fmaxf(v, 0.0f);
}

// ---- head GEMM + bias + activation ----------------------------------------
// OUT[head][10000,512] = act( AGG[10000,40] @ Wh[40,512] + bh )
// B from transposed Wht[head][512][40]. blockDim=128 (4 waves), grid=(625,8,3).
__global__ void head_gemm_wmma(const float* __restrict__ AGG,
                               const float* __restrict__ Wht,
                               const float* __restrict__ bm, const float* __restrict__ bd,
                               const float* __restrict__ bp,
                               float* __restrict__ OUT) {
  const int lane  = threadIdx.x & 31;
  const int warp  = threadIdx.x >> 5;
  const int mTile = blockIdx.x;
  const int nTile = blockIdx.y * 4 + warp;   // 0..31
  const int head  = blockIdx.z;              // 0:mean 1:disp 2:pi
  const float* B  = (head == 0) ? bm : (head == 1) ? bd : bp;
  const float* Wt = Wht + (long)head * FIN * HD;
  const int half = lane >> 4;
  const int l16  = lane & 15;
  const int col  = nTile * 16 + l16;
  const long rowA = (long)(mTile * 16 + l16) * HD;
  const long colB = (long)col * HD;

  v8f acc = {};
#pragma unroll
  for (int k0 = 0; k0 < HD; k0 += 4) {       // K = 40, exact
    const int ka = k0 + 2 * half;
    v2f a = *(const v2f*)(AGG + rowA + ka);  // AGG[row][ka..ka+1]
    v2f b = *(const v2f*)(Wt  + colB + ka);  // W[ka][col], W[ka+1][col]
    acc = wmma_f32x4(a, b, acc);
  }
  const float bias = B[col];
  float* out = OUT + (size_t)head * NN * FIN;
#pragma unroll
  for (int v = 0; v < 8; ++v) {
    int m = mTile * 16 + v + 8 * half;
    float x = acc[v] + bias;
    float r;
    if (head == 0) {                       // clip(exp(x), 1e-5, 1e6)
      r = fminf(fmaxf(expf(x), 1e-5f), 1e6f);
    } else if (head == 1) {                // clip(softplus(x), 1e-4, 1e4)
      float sp = (x > 20.0f) ? x : log1pf(expf(x));
      r = fminf(fmaxf(sp, 1e-4f), 1e4f);
    } else {                               // sigmoid
      r = 1.0f / (1.0f + expf(-x));
    }
    out[(size_t)m * FIN + col] = r;
  }
}

extern "C" void kernel_launch(void* const* d_in, const int* in_sizes, int n_in,
                              void* d_out, int out_size, void* d_ws, size_t ws_size,
                              hipStream_t stream) {
  const float* x   = (const float*)d_in[0];
  const int*   src = (const int*)  d_in[1];
  const int*   dst = (const int*)  d_in[2];
  const float* W1  = (const float*)d_in[3];
  const float* b1  = (const float*)d_in[4];
  const float* gnw = (const float*)d_in[5];
  const float* gnb = (const float*)d_in[6];
  const float* gna = (const float*)d_in[7];
  const float* Wm  = (const float*)d_in[8];
  const float* bm  = (const float*)d_in[9];
  const float* Wd  = (const float*)d_in[10];
  const float* bd  = (const float*)d_in[11];
  const float* Wp  = (const float*)d_in[12];
  const float* bp  = (const float*)d_in[13];
  const int E = in_sizes[1];

  float* ws    = (float*)d_ws;
  float* deg   = ws;                   // 10016 floats
  float* h1    = ws + 10016;           // [10000][48] = 480000
  float* agg1  = ws + 490016;          // [10000][40] = 400000 (becomes `hidden`)
  float* aggH  = ws + 890016;          // [10000][40] = 400000
  float* stats = ws + 1290016;         // 128
  float* W1t   = ws + 1290144;         // [48][512]   = 24576
  float* Wht   = ws + 1314720;         // [3][512][40]= 61440   (total ~5.5 MB)

  float* out = (float*)d_out;

  const int elemBlocks = (NN * HD + 255) / 256;
  const int edgeBlocks = (E + 255) / 256;
  const int prepBlocks = (3 * FIN * HD + 255) / 256;

  init_kernel<<<elemBlocks, 256, 0, stream>>>(deg, agg1, aggH, b1);
  prep_weights<<<prepBlocks, 256, 0, stream>>>(W1, Wm, Wd, Wp, W1t, Wht);
  degree_kernel<<<edgeBlocks, 256, 0, stream>>>(dst, deg, E);
  gemm1_wmma<<<MT, 96, 0, stream>>>(x, W1t, h1);
  scatter_kernel<<<edgeBlocks, 256, 0, stream>>>(h1, deg, src, dst, agg1, E, HP);
  gn_stats_kernel<<<HD, 256, 0, stream>>>(agg1, gnw, gnb, gna, stats);
  norm_relu_kernel<<<elemBlocks, 256, 0, stream>>>(agg1, stats);
  scatter_kernel<<<edgeBlocks, 256, 0, stream>>>(agg1, deg, src, dst, aggH, E, HD);
  head_gemm_wmma<<<dim3(MT, 8, 3), 128, 0, stream>>>(aggH, Wht, bm, bd, bp, out);
}